// Loss_40389872451982
// MI455X (gfx1250) — compile-verified
//
#include <hip/hip_runtime.h>
#include <cmath>

#define B_   32
#define G_   100
#define A_   8400
#define TPB  256
#define NB_A ((A_ + TPB - 1) / TPB)   // 33 blocks over anchors
#define NBLK3 (B_ * NB_A)             // 1056 partial slots

typedef float v2f __attribute__((ext_vector_type(2)));
typedef float v8f __attribute__((ext_vector_type(8)));

__device__ __forceinline__ float sigmoidf_(float x) { return 1.0f / (1.0f + expf(-x)); }
// numerically stable softplus == jax.nn.softplus
__device__ __forceinline__ float softplusf_(float x) { return fmaxf(x, 0.0f) + log1pf(expf(-fabsf(x))); }

__device__ __forceinline__ float iou_cxcywh(float gcx, float gcy, float gw, float gh,
                                            float px, float py, float pw, float ph) {
  float tlx = fmaxf(gcx - gw * 0.5f, px - pw * 0.5f);
  float tly = fmaxf(gcy - gh * 0.5f, py - ph * 0.5f);
  float brx = fminf(gcx + gw * 0.5f, px + pw * 0.5f);
  float bry = fminf(gcy + gh * 0.5f, py + ph * 0.5f);
  float iw = fmaxf(brx - tlx, 0.0f);
  float ih = fmaxf(bry - tly, 0.0f);
  float inter = iw * ih;
  return inter / (gw * gh + pw * ph - inter + 1e-12f);
}

// lexicographic (cost, idx) — matches stable double-argsort ranks
__device__ __forceinline__ bool pless(float c1, int i1, float c2, int i2) {
  return (c1 < c2) || (c1 == c2 && i1 < i2);
}

// ---------------- K0: zero match counters ----------------
__global__ void k0_init(int* __restrict__ cnt, int* __restrict__ mg, int n) {
  int i = blockIdx.x * blockDim.x + threadIdx.x;
  if (i < n) { cnt[i] = 0; mg[i] = 0x7fffffff; }
}

// ---------------- K1: geometric fg mask per (b,a) ----------------
__global__ void k1_fg(const float* __restrict__ outputs, const float* __restrict__ labels,
                      const float* __restrict__ xs, const float* __restrict__ ys,
                      const float* __restrict__ st, int* __restrict__ fg) {
  __shared__ float slab[G_ * 5];
  int b = blockIdx.y;
  const float* lb = labels + b * G_ * 5;
  for (int i = threadIdx.x; i < G_ * 5; i += blockDim.x) slab[i] = lb[i];
  __syncthreads();
  int a = blockIdx.x * blockDim.x + threadIdx.x;
  if (a >= A_) return;
  float s = st[a];
  float xc = (xs[a] + 0.5f) * s;
  float yc = (ys[a] + 0.5f) * s;
  float r = 2.5f * s;
  int any = 0;
  for (int g = 0; g < G_; ++g) {
    float l0 = slab[g*5+0], gcx = slab[g*5+1], gcy = slab[g*5+2], gw = slab[g*5+3], gh = slab[g*5+4];
    bool valid = (l0 + gcx + gcy + gw + gh) > 0.0f;
    bool ib = (xc > gcx - gw*0.5f) && (xc < gcx + gw*0.5f) &&
              (yc > gcy - gh*0.5f) && (yc < gcy + gh*0.5f);
    bool ic = (fabsf(xc - gcx) < r) && (fabsf(yc - gcy) < r);
    any |= (valid && (ib || ic)) ? 1 : 0;
  }
  fg[b * A_ + a] = any;
}

// ---------------- K2: one wave32 per (b,g) row: top-10 iou + top-10 (cost,idx) ----------------
__global__ void __launch_bounds__(32)
k2_rows(const float* __restrict__ outputs, const float* __restrict__ labels,
        const float* __restrict__ xs, const float* __restrict__ ys,
        const float* __restrict__ st, const int* __restrict__ fg,
        int* __restrict__ cnt, int* __restrict__ mg) {
  int row = blockIdx.x;               // b*G + g
  int b = row / G_, g = row % G_;
  int lane = threadIdx.x;             // 0..31 (wave32)
  const float* lb = labels + (b * G_ + g) * 5;
  float l0 = lb[0], gcx = lb[1], gcy = lb[2], gw = lb[3], gh = lb[4];
  bool valid = (l0 + gcx + gcy + gw + gh) > 0.0f;
  if (!valid) return;                 // invalid gt row: no candidates, no matches
  const float* ob = outputs + (size_t)b * A_ * 6;
  const int* fgb = fg + b * A_;

  float tki[10];                      // per-lane top-10 masked ious (descending)
  float tkc[10]; int tkx[10];         // per-lane top-10 (cost,idx) ascending lexicographic
#pragma unroll
  for (int i = 0; i < 10; ++i) { tki[i] = 0.0f; tkc[i] = 3.0e38f; tkx[i] = 0x7fffffff; }

  for (int a = lane; a < A_; a += 32) {
    const float* o = ob + a * 6;
    __builtin_prefetch(o + 6 * 32, 0, 0);   // gfx1250 global_prefetch for next strided row
    float px = o[0], py = o[1], pw = o[2], ph = o[3], obj = o[4], cl = o[5];
    float s = st[a];
    float xc = (xs[a] + 0.5f) * s;
    float yc = (ys[a] + 0.5f) * s;
    bool ib = (xc > gcx - gw*0.5f) && (xc < gcx + gw*0.5f) &&
              (yc > gcy - gh*0.5f) && (yc < gcy + gh*0.5f);
    float r = 2.5f * s;
    bool ic = (fabsf(xc - gcx) < r) && (fabsf(yc - gcy) < r);
    bool geo = ib && ic;
    bool ok = fgb[a] != 0;            // pair_ok (valid is known true here)
    float iou = iou_cxcywh(gcx, gcy, gw, gh, px, py, pw, ph);
    float ioum = ok ? iou : 0.0f;
    float p = sqrtf(sigmoidf_(cl) * sigmoidf_(obj));
    float cost = -logf(p + 1e-9f) - 3.0f * logf(iou + 1e-8f) + (geo ? 0.0f : 100000.0f);
    cost = ok ? cost : 1e9f;

    if (ioum > tki[9]) {
      tki[9] = ioum;
#pragma unroll
      for (int j = 9; j > 0; --j)
        if (tki[j] > tki[j-1]) { float t = tki[j]; tki[j] = tki[j-1]; tki[j-1] = t; }
    }
    if (pless(cost, a, tkc[9], tkx[9])) {
      tkc[9] = cost; tkx[9] = a;
#pragma unroll
      for (int j = 9; j > 0; --j)
        if (pless(tkc[j], tkx[j], tkc[j-1], tkx[j-1])) {
          float tc = tkc[j]; tkc[j] = tkc[j-1]; tkc[j-1] = tc;
          int   tx = tkx[j]; tkx[j] = tkx[j-1]; tkx[j-1] = tx;
        }
    }
  }

  __shared__ float s_iou[320];
  __shared__ float s_c[320];
  __shared__ int   s_x[320];
#pragma unroll
  for (int i = 0; i < 10; ++i) {
    s_iou[lane * 10 + i] = tki[i];
    s_c[lane * 10 + i] = tkc[i];
    s_x[lane * 10 + i] = tkx[i];
  }
  __syncthreads();
  if (lane == 0) {
    float mi[10]; float mc[10]; int mx[10];
    for (int i = 0; i < 10; ++i) { mi[i] = 0.0f; mc[i] = 3.0e38f; mx[i] = 0x7fffffff; }
    for (int i = 0; i < 320; ++i) {
      float v = s_iou[i];
      if (v > mi[9]) {
        mi[9] = v;
        for (int j = 9; j > 0; --j)
          if (mi[j] > mi[j-1]) { float t = mi[j]; mi[j] = mi[j-1]; mi[j-1] = t; }
      }
      float cv = s_c[i]; int ci = s_x[i];
      if (pless(cv, ci, mc[9], mx[9])) {
        mc[9] = cv; mx[9] = ci;
        for (int j = 9; j > 0; --j)
          if (pless(mc[j], mx[j], mc[j-1], mx[j-1])) {
            float t = mc[j]; mc[j] = mc[j-1]; mc[j-1] = t;
            int   x = mx[j]; mx[j] = mx[j-1]; mx[j-1] = x;
          }
      }
    }
    float ssum = 0.0f;
    for (int i = 0; i < 10; ++i) ssum += mi[i];
    int dynk = (int)ssum;             // truncation, matches .astype(int32)
    if (dynk < 1) dynk = 1;
    if (dynk > 10) dynk = 10;
    for (int i = 0; i < dynk; ++i) {
      if (mc[i] < 1e9f) {             // cost<1e9 <=> pair_ok
        atomicAdd(&cnt[b * A_ + mx[i]], 1);
        atomicMin(&mg[b * A_ + mx[i]], g);
      }
    }
  }
}

// ---------------- K3: per-anchor resolution + deterministic block reduction ----------------
__global__ void k3_resolve(const float* __restrict__ outputs, const float* __restrict__ labels,
                           const float* __restrict__ xs, const float* __restrict__ ys,
                           const float* __restrict__ st,
                           const int* __restrict__ cnt, const int* __restrict__ mg,
                           float* __restrict__ partials) {
  __shared__ float slab[G_ * 5];
  __shared__ float r1[TPB];
  __shared__ float r2[TPB];
  int b = blockIdx.y;
  const float* lb = labels + b * G_ * 5;
  for (int i = threadIdx.x; i < G_ * 5; i += blockDim.x) slab[i] = lb[i];
  __syncthreads();
  int a = blockIdx.x * blockDim.x + threadIdx.x;
  float contrib = 0.0f, fgf = 0.0f;
  if (a < A_) {
    int c = cnt[b * A_ + a];
    if (c > 0) {
      const float* o = outputs + ((size_t)b * A_ + a) * 6;
      float px = o[0], py = o[1], pw = o[2], ph = o[3], obj = o[4], cl = o[5];
      int gsel;
      if (c == 1) {
        gsel = mg[b * A_ + a];        // the unique matching gt (min g)
      } else {
        // cnt>1: keep argmin-cost gt over the full column (first tie wins)
        float s = st[a];
        float xc = (xs[a] + 0.5f) * s, yc = (ys[a] + 0.5f) * s, r = 2.5f * s;
        float p = sqrtf(sigmoidf_(cl) * sigmoidf_(obj));
        float ccst = -logf(p + 1e-9f);
        float best = 3.0e38f; int bg = 0;
        for (int g = 0; g < G_; ++g) {
          float l0 = slab[g*5+0], gcx = slab[g*5+1], gcy = slab[g*5+2], gw = slab[g*5+3], gh = slab[g*5+4];
          bool vg = (l0 + gcx + gcy + gw + gh) > 0.0f;
          float cost;
          if (vg) {  // fg[a] is true here since this anchor matched
            bool ib = (xc > gcx - gw*0.5f) && (xc < gcx + gw*0.5f) &&
                      (yc > gcy - gh*0.5f) && (yc < gcy + gh*0.5f);
            bool ic = (fabsf(xc - gcx) < r) && (fabsf(yc - gcy) < r);
            float iou = iou_cxcywh(gcx, gcy, gw, gh, px, py, pw, ph);
            cost = ccst - 3.0f * logf(iou + 1e-8f) + ((ib && ic) ? 0.0f : 100000.0f);
          } else {
            cost = 1e9f;
          }
          if (cost < best) { best = cost; bg = g; }
        }
        gsel = bg;
      }
      float gcx = slab[gsel*5+1], gcy = slab[gsel*5+2], gw = slab[gsel*5+3], gh = slab[gsel*5+4];
      float pred_iou = iou_cxcywh(gcx, gcy, gw, gh, px, py, pw, ph);
      float cls_t = pred_iou;         // one_hot(0,1)==1 always (single class, label 0)
      float bce = cls_t * softplusf_(-cl) + (1.0f - cls_t) * softplusf_(cl);
      contrib = bce; fgf = 1.0f;
    }
  }
  r1[threadIdx.x] = contrib; r2[threadIdx.x] = fgf;
  __syncthreads();
  for (int sft = TPB / 2; sft > 0; sft >>= 1) {   // fixed-order => deterministic
    if (threadIdx.x < sft) {
      r1[threadIdx.x] += r1[threadIdx.x + sft];
      r2[threadIdx.x] += r2[threadIdx.x + sft];
    }
    __syncthreads();
  }
  if (threadIdx.x == 0) {
    int bid = blockIdx.y * gridDim.x + blockIdx.x;
    partials[bid] = r1[0];
    partials[NBLK3 + bid] = r2[0];
  }
}

// ---------------- K4: final reduce (single wave32) with ones-matrix WMMA ----------------
// D = ones(16x4) x B + 0 gives column sums of B; summing D's VGPR0 across the
// 32 lanes yields 2 * sum(all 64 B slots) per the documented 16x16 f32 C/D
// layout (lanes 0-15: M=0 row, lanes 16-31: M=8 row; all rows identical).
__global__ void __launch_bounds__(32)
k4_final(const float* __restrict__ partials, float* __restrict__ out) {
  int lane = threadIdx.x;
  float s1 = 0.0f, s2 = 0.0f;
  for (int i = lane; i < NBLK3; i += 32) {
    s1 += partials[i];
    s2 += partials[NBLK3 + i];
  }
  v2f aones; aones.x = 1.0f; aones.y = 1.0f;      // all-ones A covers all 64 A slots
  v8f cz = {0.f, 0.f, 0.f, 0.f, 0.f, 0.f, 0.f, 0.f};
  v2f b1; b1.x = s1; b1.y = 0.0f;
  v2f b2; b2.x = s2; b2.y = 0.0f;
  v8f d1 = __builtin_amdgcn_wmma_f32_16x16x4_f32(false, aones, false, b1, (short)0, cz, false, false);
  v8f d2 = __builtin_amdgcn_wmma_f32_16x16x4_f32(false, aones, false, b2, (short)0, cz, false, false);
  float t1 = d1[0], t2 = d2[0];
#pragma unroll
  for (int m = 16; m >= 1; m >>= 1) {
    t1 += __shfl_xor(t1, m, 32);
    t2 += __shfl_xor(t2, m, 32);
  }
  if (lane == 0) {
    float loss_sum = 0.5f * t1;       // undo the 2x duplication (rows M=0 and M=8)
    float num_fg   = 0.5f * t2;
    out[0] = loss_sum / fmaxf(num_fg, 1.0f);
  }
}

extern "C" void kernel_launch(void* const* d_in, const int* in_sizes, int n_in,
                              void* d_out, int out_size, void* d_ws, size_t ws_size,
                              hipStream_t stream) {
  (void)in_sizes; (void)n_in; (void)out_size; (void)ws_size;
  const float* outputs = (const float*)d_in[0];   // (32, 8400, 6)
  const float* labels  = (const float*)d_in[1];   // (32, 100, 5)
  const float* xs      = (const float*)d_in[2];   // (1, 8400)
  const float* ys      = (const float*)d_in[3];   // (1, 8400)
  const float* st      = (const float*)d_in[4];   // (1, 8400)
  float* out = (float*)d_out;

  char* ws = (char*)d_ws;
  int*   fg       = (int*)ws;                                    // B*A ints
  int*   cnt      = (int*)(ws + 1 * sizeof(int) * B_ * A_);      // B*A ints
  int*   mg       = (int*)(ws + 2 * sizeof(int) * B_ * A_);      // B*A ints
  float* partials = (float*)(ws + 3 * sizeof(int) * B_ * A_);    // 2*NBLK3 floats

  int n = B_ * A_;
  k0_init<<<(n + 255) / 256, 256, 0, stream>>>(cnt, mg, n);
  k1_fg<<<dim3(NB_A, B_), TPB, 0, stream>>>(outputs, labels, xs, ys, st, fg);
  k2_rows<<<B_ * G_, 32, 0, stream>>>(outputs, labels, xs, ys, st, fg, cnt, mg);
  k3_resolve<<<dim3(NB_A, B_), TPB, 0, stream>>>(outputs, labels, xs, ys, st, cnt, mg, partials);
  k4_final<<<1, 32, 0, stream>>>(partials, out);
}